// InsectBrain_15195594293735
// MI455X (gfx1250) — compile-verified
//
#include <hip/hip_runtime.h>
#include <math.h>

// ---------------- problem constants ----------------
#define B_   256
#define T_   512
#define IN_  16
#define H_   128
#define TOT_ 124
#define BT_  (B_ * T_)

typedef float v2f __attribute__((ext_vector_type(2)));
typedef float v8f __attribute__((ext_vector_type(8)));

#if defined(__has_builtin)
#if __has_builtin(__builtin_amdgcn_wmma_f32_16x16x4_f32)
#define HAVE_WMMA_F32X4 1
#endif
#endif

// Flattened input pointers. Assumption: harness flattens the setup_inputs()
// dict depth-first in insertion order:
//  0: x
//  1,2: input_proj.w/.b
//  3+7*rg .. : region rg leaves (in.w, in.b, hh.w, hh.b, log_step, ln_g, ln_b)
//      regions in order: antennal, optic, mb_calyx, mb_lobes, lateral_horn,
//                        central_cx, protocerebrum, seg
//  59..62: mb_to_antennal (proj.w, proj.b, gate.w, gate.b)
//  63..66: lh_inhibit_mb  (proj.w, proj.b, gate.w, gate.b)
//  67,68: thalamus_lin.w/.b   69,70: thalamus_ln_g/_b
//  71,72: thalamic_proj.w/.b  73,74: next_ln_g/_b  75,76: next_lin.w/.b
//  77,78: anom_ln_g/_b  79,80: anom_lin.w/.b
//  81,82: prop_ln_g/_b  83,84: prop_lin1.w/.b  85,86: prop_lin2.w/.b
struct Params { const float* p[87]; };

struct Outs {
  float* preds;   // [B, T-1, 16]
  float* hidden;  // [B, T, 128]
  float* anom;    // [B, T]
  float* surp;    // [B, T-1]
  float* props;   // [B, 2]
  float* tr[8];   // traces, [B, T, nh] each
};

// region meta (state-vector offsets match concat order => all_r == S[0..124))
static __device__ const int   R_OFF[8] = {0, 12, 27, 46, 65, 77, 96, 112};
static __device__ const int   R_NH [8] = {12, 15, 19, 19, 12, 19, 16, 12};
static __device__ const int   R_EFF[8] = {12, 12, 27, 38, 27, 50, 66, 35};
static __device__ const int   R_WIN[8] = {0, 144, 324, 837, 1559, 1883, 2833, 3889};
static __device__ const int   R_WHH[8] = {0, 144, 369, 730, 1091, 1235, 1596, 1852};
static __device__ const float R_DT [8] = {2.5f, 2.5f, 1.0f, 0.5f, 2.0f, 0.8f, 0.6f, 1.8f};

__device__ __forceinline__ int region_of(int k) {
  if (k < 12)  return 0;
  if (k < 27)  return 1;
  if (k < 46)  return 2;
  if (k < 65)  return 3;
  if (k < 77)  return 4;
  if (k < 96)  return 5;
  if (k < 112) return 6;
  return 7;
}

// =====================================================================
// Kernel A: the serial recurrence. ONE batch row per single-wave
// workgroup -> 256 independent workgroups, every region update is at
// most 19 parallel dot products == at most one dot per lane on the
// 512-step critical path. All weights LDS-resident; __syncthreads() in
// a single-wave workgroup lowers to s_wait_dscnt only (s_barrier->NOP).
// =====================================================================
__global__ __launch_bounds__(32) void brain_scan_kernel(Params P, Outs O)
{
  __shared__ float W_IN[4309];
  __shared__ float W_HH[1996];
  __shared__ float B_INs[124], B_HHs[124], STEPF[124], LNG[124], LNB[124];
  __shared__ float IPW[192], IPB[12];
  __shared__ float A1PW[228], A1PB[12], A1GW[228], A1GB[12];
  __shared__ float A2PW[324], A2PB[27], A2GW[324], A2GB[27];
  __shared__ float S[124];     // region states (== all_r for this row)
  __shared__ float XIN[66];    // staging for concatenated/biased inputs
  __shared__ float HN[19];     // pre-LN new state
  __shared__ float MVS[2];     // mean / rstd
  __shared__ float XT[16];
  __shared__ float AXO1[12];
  __shared__ float AXO2[27];

  const int lane = threadIdx.x;

  auto cp = [&](float* d, const float* s, int n) {
    for (int i = lane; i < n; i += 32) d[i] = s[i];
  };

  cp(IPW, P.p[1], 192);
  cp(IPB, P.p[2], 12);
  for (int rg = 0; rg < 8; ++rg) {
    const int base = 3 + 7 * rg;
    cp(W_IN + R_WIN[rg], P.p[base + 0], R_EFF[rg] * R_NH[rg]);
    cp(B_INs + R_OFF[rg], P.p[base + 1], R_NH[rg]);
    cp(W_HH + R_WHH[rg], P.p[base + 2], R_NH[rg] * R_NH[rg]);
    cp(B_HHs + R_OFF[rg], P.p[base + 3], R_NH[rg]);
    for (int i = lane; i < R_NH[rg]; i += 32)
      STEPF[R_OFF[rg] + i] = R_DT[rg] * expf(P.p[base + 4][i]);  // dt*exp(log_step)
    cp(LNG + R_OFF[rg], P.p[base + 5], R_NH[rg]);
    cp(LNB + R_OFF[rg], P.p[base + 6], R_NH[rg]);
  }
  cp(A1PW, P.p[59], 228); cp(A1PB, P.p[60], 12);
  cp(A1GW, P.p[61], 228); cp(A1GB, P.p[62], 12);
  cp(A2PW, P.p[63], 324); cp(A2PB, P.p[64], 27);
  cp(A2GW, P.p[65], 324); cp(A2GB, P.p[66], 27);
  for (int i = lane; i < 124; i += 32) S[i] = 0.0f;
  __syncthreads();

  // LTC region step: pre = xin@Win + bin + h@Whh + bhh;
  // h' = LN(h + dt*exp(log_step)*(tanh(pre)-h))
  auto region_step = [&](int rg, const float* xin) {
    const int nh = R_NH[rg], eff = R_EFF[rg], off = R_OFF[rg];
    const float* win = W_IN + R_WIN[rg];
    const float* whh = W_HH + R_WHH[rg];
    if (lane < nh) {
      const int j = lane;
      float acc = B_INs[off + j] + B_HHs[off + j];
      for (int k = 0; k < eff; ++k) acc += xin[k] * win[k * nh + j];
      for (int k = 0; k < nh; ++k)  acc += S[off + k] * whh[k * nh + j];
      const float h = S[off + j];
      HN[j] = h + STEPF[off + j] * (tanhf(acc) - h);
    }
    __syncthreads();
    if (lane == 0) {
      float m = 0.0f;
      for (int j = 0; j < nh; ++j) m += HN[j];
      m /= (float)nh;
      float v = 0.0f;
      for (int j = 0; j < nh; ++j) { const float d = HN[j] - m; v += d * d; }
      MVS[0] = m;
      MVS[1] = rsqrtf(v / (float)nh + 1e-5f);
    }
    __syncthreads();
    if (lane < nh)
      S[off + lane] = (HN[lane] - MVS[0]) * MVS[1] * LNG[off + lane] + LNB[off + lane];
    __syncthreads();
  };

  const float* x = P.p[0];
  const int b = blockIdx.x;

  for (int t = 0; t < T_; ++t) {
    const float tm = (t > 0) ? 0.1f : 0.0f;  // 0.1 * tmask

    // combined phase: x_t gather + both axons (from carry: mbl_old@S+46, lh_old@S+65)
    for (int task = lane; task < 55; task += 32) {
      if (task < 16) {
        XT[task] = x[((size_t)b * T_ + t) * IN_ + task];
      } else if (task < 28) {            // mb_to_antennal: 19 -> 12
        const int j = task - 16;
        float pg = A1GB[j], pp = A1PB[j];
        for (int k = 0; k < 19; ++k) {
          const float m = S[46 + k];
          pg += m * A1GW[k * 12 + j];
          pp += m * A1PW[k * 12 + j];
        }
        AXO1[j] = tanhf(pp) / (1.0f + expf(-pg));
      } else {                           // lh_inhibit_mb: 12 -> 27
        const int j = task - 28;
        float pg = A2GB[j], pp = A2PB[j];
        for (int k = 0; k < 12; ++k) {
          const float m = S[65 + k];
          pg += m * A2GW[k * 27 + j];
          pp += m * A2PW[k * 27 + j];
        }
        AXO2[j] = tanhf(pp) / (1.0f + expf(-pg));
      }
    }
    __syncthreads();

    // antennal input = input_proj(x_t) + 0.1*tmask*axon1
    if (lane < 12) {
      float acc = IPB[lane];
      for (int k = 0; k < 16; ++k) acc += XT[k] * IPW[k * 12 + lane];
      XIN[lane] = acc + tm * AXO1[lane];
    }
    __syncthreads();
    region_step(0, XIN);       // antennal
    region_step(1, S + 0);     // optic (in = ant_new)

    // mb_calyx input = concat(ant,opt) + 0.1*tmask*axon2
    if (lane < 27) XIN[lane] = S[lane] + tm * AXO2[lane];
    __syncthreads();
    region_step(2, XIN);       // mb_calyx
    region_step(3, S + 27);    // mb_lobes (concat(mbc_new, mbl_old), self)
    region_step(4, S + 0);     // lateral_horn (concat(ant,opt))
    region_step(5, S + 46);    // central_cx (concat(mbl,lh,cc_old), self)
    region_step(6, S + 46);    // protocerebrum (concat(mbl,lh,cc,proto_old), self)

    // seg input = concat(proto, cc)  (non-contiguous in S)
    for (int task = lane; task < 35; task += 32)
      XIN[task] = (task < 16) ? S[96 + task] : S[77 + (task - 16)];
    __syncthreads();
    region_step(7, XIN);       // seg

    // write traces (== per-region slices of S)
    for (int rg = 0; rg < 8; ++rg) {
      const int nh = R_NH[rg], off = R_OFF[rg];
      if (lane < nh)
        O.tr[rg][((size_t)b * T_ + t) * nh + lane] = S[off + lane];
    }
    __syncthreads();
  }
}

// =====================================================================
// Kernel B: thalamus gate+proj over all B*T rows — the FLOP-heavy,
// fully parallel part. f32 WMMA (V_WMMA_F32_16X16X4_F32): two 16-row
// M-tiles per block (32 rows), K = 124 = 31*4, 8 waves per block each
// owning one 16-col N-tile. Every loaded B fragment feeds 2 A fragments
// x 2 weight matrices -> 4 WMMA per k-step per wave, halving the L2
// weight traffic vs 1 M-tile. Weights (127 KB) are prefetched and stay
// L2-resident for all 4096 blocks. LN + sigmoid gating fused via LDS.
// =====================================================================
__global__ __launch_bounds__(256) void thalamus_wmma_kernel(Params P, Outs O)
{
  __shared__ float As[32][124];
  __shared__ float Us[32][128];
  __shared__ float Ps[32][128];
  __shared__ float MU[32], RS[32];

  const int tid = threadIdx.x;
  const int r0 = blockIdx.x * 32;  // global row = b*T + t

  const float* __restrict__ W1 = P.p[67];  // thalamus_lin.w  [124,128]
  const float* __restrict__ W2 = P.p[71];  // thalamic_proj.w [124,128]

  // warm L2 with both weight matrices (256 threads x 496B apart covers 127KB)
  __builtin_prefetch(W1 + (size_t)tid * 124, 0, 1);
  __builtin_prefetch(W2 + (size_t)tid * 124, 0, 1);

  // stage A tiles by gathering the 8 region traces (concat order == offsets)
  for (int i = tid; i < 32 * 124; i += 256) {
    const int row = i / 124, k = i - row * 124;
    const int rg = region_of(k);
    As[row][k] = O.tr[rg][(size_t)(r0 + row) * R_NH[rg] + (k - R_OFF[rg])];
  }
  __syncthreads();

  const int lane = tid & 31;
  const int wv = tid >> 5;
  const int n0 = wv * 16 + (lane & 15);   // global output column
  const int kb = (lane >> 4) * 2;         // K sub-offset for A/B fragments
  const int m = lane & 15;                // A row for this lane
  const int half = lane >> 4;

  v8f c1a = {0.f, 0.f, 0.f, 0.f, 0.f, 0.f, 0.f, 0.f};
  v8f c1b = c1a, c2a = c1a, c2b = c1a;

#if defined(HAVE_WMMA_F32X4)
  for (int k0 = 0; k0 < 124; k0 += 4) {
    v2f a0; a0.x = As[m][k0 + kb];        a0.y = As[m][k0 + kb + 1];
    v2f a1; a1.x = As[16 + m][k0 + kb];   a1.y = As[16 + m][k0 + kb + 1];
    v2f bA; bA.x = W1[(k0 + kb) * H_ + n0]; bA.y = W1[(k0 + kb + 1) * H_ + n0];
    v2f bB; bB.x = W2[(k0 + kb) * H_ + n0]; bB.y = W2[(k0 + kb + 1) * H_ + n0];
    c1a = __builtin_amdgcn_wmma_f32_16x16x4_f32(false, a0, false, bA, (short)0, c1a, false, false);
    c1b = __builtin_amdgcn_wmma_f32_16x16x4_f32(false, a1, false, bA, (short)0, c1b, false, false);
    c2a = __builtin_amdgcn_wmma_f32_16x16x4_f32(false, a0, false, bB, (short)0, c2a, false, false);
    c2b = __builtin_amdgcn_wmma_f32_16x16x4_f32(false, a1, false, bB, (short)0, c2b, false, false);
  }
#else
  // VALU fallback with identical C/D lane layout (also used by the host pass)
  for (int v = 0; v < 8; ++v) {
    const int mr = v + 8 * half;
    float s1a = 0.f, s1b = 0.f, s2a = 0.f, s2b = 0.f;
    for (int k = 0; k < 124; ++k) {
      const float w1 = W1[k * H_ + n0], w2 = W2[k * H_ + n0];
      s1a += As[mr][k] * w1;      s2a += As[mr][k] * w2;
      s1b += As[16 + mr][k] * w1; s2b += As[16 + mr][k] * w2;
    }
    c1a[v] = s1a; c1b[v] = s1b; c2a[v] = s2a; c2b[v] = s2b;
  }
#endif

  const float* b1 = P.p[68];
  const float* b2 = P.p[72];
  for (int v = 0; v < 8; ++v) {
    const int mr = v + 8 * half;  // C/D layout: M = v + 8*(lane>>4), N = lane&15
    Us[mr][n0]      = c1a[v] + b1[n0];
    Ps[mr][n0]      = c2a[v] + b2[n0];
    Us[16 + mr][n0] = c1b[v] + b1[n0];
    Ps[16 + mr][n0] = c2b[v] + b2[n0];
  }
  __syncthreads();

  if (tid < 32) {
    float mn = 0.0f;
    for (int n = 0; n < H_; ++n) mn += Us[tid][n];
    mn *= (1.0f / H_);
    float vv = 0.0f;
    for (int n = 0; n < H_; ++n) { const float d = Us[tid][n] - mn; vv += d * d; }
    MU[tid] = mn;
    RS[tid] = rsqrtf(vv * (1.0f / H_) + 1e-5f);
  }
  __syncthreads();

  const float* lg = P.p[69];
  const float* lb = P.p[70];
  for (int i = tid; i < 32 * H_; i += 256) {
    const int row = i >> 7, n = i & 127;
    const float u = (Us[row][n] - MU[row]) * RS[row] * lg[n] + lb[n];
    const float g = 1.0f / (1.0f + expf(-u));
    O.hidden[(size_t)(r0 + row) * H_ + n] = g * Ps[row][n];
  }
}

// =====================================================================
// Kernel C: heads over hidden — preds (+surprise) and anomaly.
// One thread per (b,t) row; row (512 B) hoisted into registers once.
// =====================================================================
__global__ __launch_bounds__(256) void heads_kernel(Params P, Outs O)
{
  const int r = blockIdx.x * 256 + threadIdx.x;
  if (r >= BT_) return;
  const float* __restrict__ h = O.hidden + (size_t)r * H_;

  float hv[H_];
#pragma unroll
  for (int k = 0; k < H_; ++k) hv[k] = h[k];

  float s = 0.0f, s2 = 0.0f;
#pragma unroll
  for (int k = 0; k < H_; ++k) { s += hv[k]; s2 += hv[k] * hv[k]; }
  const float mn = s * (1.0f / H_);
  const float rs = rsqrtf(s2 * (1.0f / H_) - mn * mn + 1e-5f);

  {  // anomaly = sigmoid(LN(h; anom) @ aw + ab)
    const float* ag = P.p[77]; const float* ab = P.p[78]; const float* aw = P.p[79];
    float acc = P.p[80][0];
#pragma unroll 8
    for (int k = 0; k < H_; ++k) acc += ((hv[k] - mn) * rs * ag[k] + ab[k]) * aw[k];
    O.anom[r] = 1.0f / (1.0f + expf(-acc));
  }

  const int b = r >> 9;          // T_ == 512
  const int t = r & (T_ - 1);
  if (t < T_ - 1) {              // preds + surprise use hidden[:, :-1]
    const float* ng = P.p[73]; const float* nb = P.p[74];
    const float* nw = P.p[75]; const float* nbb = P.p[76];
    float pj[16];
    for (int j = 0; j < 16; ++j) pj[j] = nbb[j];
#pragma unroll 4
    for (int k = 0; k < H_; ++k) {
      const float hn = (hv[k] - mn) * rs * ng[k] + nb[k];
      for (int j = 0; j < 16; ++j) pj[j] += hn * nw[k * 16 + j];
    }
    const size_t po = ((size_t)b * (T_ - 1) + t) * 16;
    const float* xn = P.p[0] + ((size_t)b * T_ + t + 1) * IN_;
    float se = 0.0f;
    for (int j = 0; j < 16; ++j) {
      O.preds[po + j] = pj[j];
      const float d = pj[j] - xn[j];
      se += d * d;
    }
    O.surp[(size_t)b * (T_ - 1) + t] = se * (1.0f / 16.0f);
  }
}

// =====================================================================
// Kernel D: props head on hidden[:, -1]  (256 rows; 2-layer MLP, exact GELU)
// =====================================================================
__global__ __launch_bounds__(256) void props_kernel(Params P, Outs O)
{
  const int b = threadIdx.x;
  if (b >= B_) return;
  const float* __restrict__ h = O.hidden + ((size_t)b * T_ + (T_ - 1)) * H_;

  float s = 0.0f, s2 = 0.0f;
  for (int k = 0; k < H_; ++k) { const float v = h[k]; s += v; s2 += v * v; }
  const float mn = s * (1.0f / H_);
  const float rs = rsqrtf(s2 * (1.0f / H_) - mn * mn + 1e-5f);

  const float* g = P.p[81]; const float* gb = P.p[82];
  const float* w1 = P.p[83]; const float* b1 = P.p[84];
  float z[64];
  for (int j = 0; j < 64; ++j) z[j] = b1[j];
  for (int k = 0; k < H_; ++k) {
    const float hn = (h[k] - mn) * rs * g[k] + gb[k];
    for (int j = 0; j < 64; ++j) z[j] += hn * w1[k * 64 + j];
  }
  const float* w2 = P.p[85]; const float* b2 = P.p[86];
  float o0 = b2[0], o1 = b2[1];
  for (int j = 0; j < 64; ++j) {
    const float v = z[j];
    const float gel = 0.5f * v * (1.0f + erff(v * 0.70710678118f));  // exact GELU
    o0 += gel * w2[j * 2 + 0];
    o1 += gel * w2[j * 2 + 1];
  }
  O.props[(size_t)b * 2 + 0] = o0;
  O.props[(size_t)b * 2 + 1] = o1;
}

// =====================================================================
extern "C" void kernel_launch(void* const* d_in, const int* in_sizes, int n_in,
                              void* d_out, int out_size, void* d_ws, size_t ws_size,
                              hipStream_t stream)
{
  (void)in_sizes; (void)out_size; (void)d_ws; (void)ws_size;

  Params P;
  for (int i = 0; i < 87; ++i)
    P.p[i] = (i < n_in) ? (const float*)d_in[i] : (const float*)d_in[0];

  Outs O;
  float* o = (float*)d_out;
  O.preds  = o; o += (size_t)B_ * (T_ - 1) * IN_;
  O.hidden = o; o += (size_t)B_ * T_ * H_;
  O.anom   = o; o += (size_t)B_ * T_;
  O.surp   = o; o += (size_t)B_ * (T_ - 1);
  O.props  = o; o += (size_t)B_ * 2;
  const int nhs[8] = {12, 15, 19, 19, 12, 19, 16, 12};
  for (int rg = 0; rg < 8; ++rg) { O.tr[rg] = o; o += (size_t)B_ * T_ * nhs[rg]; }

  // A: serial recurrence (256 independent single-wave workgroups)
  brain_scan_kernel<<<dim3(B_), dim3(32), 0, stream>>>(P, O);
  // B: thalamus gate/proj GEMM via f32 WMMA over all B*T rows (32 rows/block)
  thalamus_wmma_kernel<<<dim3(BT_ / 32), dim3(256), 0, stream>>>(P, O);
  // C: preds / surprise / anomaly heads
  heads_kernel<<<dim3((BT_ + 255) / 256), dim3(256), 0, stream>>>(P, O);
  // D: props head
  props_kernel<<<dim3(1), dim3(256), 0, stream>>>(P, O);
}